// ConstraintProblog_23553600651580
// MI455X (gfx1250) — compile-verified
//
#include <hip/hip_runtime.h>
#include <cstdint>
#include <cstddef>

// Problem constants (from reference)
#define V_ 4
#define D_ 10
#define B_ 4096
#define L_ 1024
#define N_ 4096
#define P_ 8
#define EPSF 1e-12f

// Tuning
#define C_ 12            // batch columns per workgroup (LDS-limited)
#define CB_ (C_ * 4)     // row stride in bytes
#define THREADS_ 128     // 4 waves (wave32)

// Dynamic LDS layout (bytes):
//   [0, 65536)              int4 idx[N_]              {gate, left, right, pad}
//   [65536, 311296)         float vals[(L_+N_)][C_]   per-column value scratch
//   [311296, 313216)        float pl[V_*D_][C_]       per-column softmax probs
#define IDX_OFF   0
#define VALS_OFF  (N_ * 16)
#define PL_OFF    (VALS_OFF + (L_ + N_) * CB_)
#define SMEM_BYTES (PL_OFF + V_ * D_ * CB_)
static_assert(SMEM_BYTES <= 320 * 1024, "exceeds 320KB WGP LDS");

// CDNA5 async global->LDS copy (ASYNCcnt path). VDST operand is the LDS byte
// offset in a VGPR; GV addressing (64-bit vaddr pair), per CDNA5 ISA.
__device__ __forceinline__ void async_g2l_b32(uint32_t lds_byte_off, const void* gaddr) {
    asm volatile("global_load_async_to_lds_b32 %0, %1, off"
                 :
                 : "v"(lds_byte_off), "v"((unsigned long long)(uintptr_t)gaddr)
                 : "memory");
}

__device__ __forceinline__ void wait_asynccnt0() {
    asm volatile("s_wait_asynccnt 0" ::: "memory");
}

__launch_bounds__(THREADS_)
__global__ void problog_scan_kernel(const float* __restrict__ imgs,
                                    const float* __restrict__ ctl,   // classes_temp_logits [V]
                                    const float* __restrict__ ptl,   // props_temp_logits   [P]
                                    const int* __restrict__ leaf_var,
                                    const int* __restrict__ leaf_cls,
                                    const int* __restrict__ leaf_neg,
                                    const int* __restrict__ gate,
                                    const int* __restrict__ nleft,
                                    const int* __restrict__ nright,
                                    float* __restrict__ out)
{
    extern __shared__ char smem[];
    int4*  idx  = reinterpret_cast<int4*>(smem + IDX_OFF);
    float* vals = reinterpret_cast<float*>(smem + VALS_OFF);
    float* pl   = reinterpret_cast<float*>(smem + PL_OFF);

    const int tid = threadIdx.x;
    const int b0  = blockIdx.x * C_;

    // ---- Phase 1: kick off async DMA of node index triples into LDS,
    //      interleaved as int4 {gate, left, right, _} for one b128 load/node.
    for (int i = tid; i < N_; i += THREADS_) {
        uint32_t lo = (uint32_t)(uintptr_t)(void*)&idx[i];  // LDS byte offset = low 32 bits
        async_g2l_b32(lo + 0, gate   + i);
        async_g2l_b32(lo + 4, nleft  + i);
        async_g2l_b32(lo + 8, nright + i);
    }

    // ---- Phase 2 (overlaps the DMA): per-(v, column) softmax over D=10.
    if (tid < V_ * C_) {
        const int v = tid / C_;
        const int c = tid % C_;
        const int b = b0 + c;
        if (b < B_) {
            const float t    = fmaxf(ctl[v], 0.0f) + EPSF;
            const float invt = 1.0f / t;
            const float* row = imgs + ((size_t)v * B_ + (size_t)b) * D_;
            float y[D_];
            float m = -INFINITY;
            #pragma unroll
            for (int d = 0; d < D_; ++d) { y[d] = row[d] * invt; m = fmaxf(m, y[d]); }
            float s = 0.0f;
            #pragma unroll
            for (int d = 0; d < D_; ++d) { y[d] = expf(y[d] - m); s += y[d]; }
            const float invs = 1.0f / s;
            #pragma unroll
            for (int d = 0; d < D_; ++d) pl[(v * D_ + d) * C_ + c] = y[d] * invs;
        }
    }

    wait_asynccnt0();     // issuing wave's async writes to LDS complete
    __syncthreads();      // make idx + pl visible workgroup-wide

    // ---- Phase 3a: rescale indices in-place into absolute LDS byte offsets
    //      (row * 48 + VALS_OFF). Removes the multiply from the scan's
    //      dependent chain; adding the lane offset (c*4) is a single v_add.
    for (int i = tid; i < N_; i += THREADS_) {
        int4 t = idx[i];
        t.x = t.x * CB_ + VALS_OFF;
        t.y = t.y * CB_ + VALS_OFF;
        t.z = t.z * CB_ + VALS_OFF;
        idx[i] = t;
    }

    // ---- Phase 3b: gather leaf probabilities into vals[0..L).
    for (int k = tid; k < L_ * C_; k += THREADS_) {
        const int l = k / C_;
        const int c = k % C_;
        float p = pl[(leaf_var[l] * D_ + leaf_cls[l]) * C_ + c];
        if (leaf_neg[l]) p = 1.0f - p;
        vals[l * C_ + c] = p;
    }
    __syncthreads();

    // ---- Phase 4: the sequential node scan. One lane per batch column;
    //      entire dependent chain lives in LDS. Index load for i+1 is
    //      software-pipelined ahead of the dependent value loads; indices are
    //      pre-scaled byte offsets so each operand address is one add.
    if (tid < C_) {
        const uint32_t c4 = (uint32_t)tid * 4u;
        char* base = smem;
        uint32_t sto = VALS_OFF + (uint32_t)L_ * CB_ + c4;
        int4 t = idx[0];
        #pragma unroll 4
        for (int i = 0; i < N_; ++i) {
            const int4 tn = idx[(i + 1 < N_) ? (i + 1) : i];
            const float g = *(const float*)(base + ((uint32_t)t.x + c4));
            const float a = *(const float*)(base + ((uint32_t)t.y + c4));
            const float b = *(const float*)(base + ((uint32_t)t.z + c4));
            *(float*)(base + sto) = fmaf(g, a, (1.0f - g) * b);
            sto += CB_;
            t = tn;
        }

        // ---- Phase 5: epilogue on the last P rows (same lanes own the data,
        //      no barrier needed).
        const int b = b0 + tid;
        if (b < B_) {
            const float* vc = vals + tid;
            #pragma unroll
            for (int p = 0; p < P_; ++p) {
                float q = vc[(L_ + N_ - P_ + p) * C_];
                q = fminf(fmaxf(q, EPSF), 1.0f - EPSF);
                const float z  = logf(q / (1.0f - q));
                const float tp = fmaxf(ptl[p], 0.0f) + EPSF;
                const float r  = z / tp;
                out[(size_t)p * B_ + b] = 1.0f / (1.0f + expf(-r));
            }
        }
    }
}

extern "C" void kernel_launch(void* const* d_in, const int* in_sizes, int n_in,
                              void* d_out, int out_size, void* d_ws, size_t ws_size,
                              hipStream_t stream) {
    (void)in_sizes; (void)n_in; (void)out_size; (void)d_ws; (void)ws_size;

    const float* imgs     = (const float*)d_in[0];
    const float* ctl      = (const float*)d_in[1];
    const float* ptl      = (const float*)d_in[2];
    const int*   leaf_var = (const int*)d_in[3];
    const int*   leaf_cls = (const int*)d_in[4];
    const int*   leaf_neg = (const int*)d_in[5];
    const int*   gate     = (const int*)d_in[6];
    const int*   nleft    = (const int*)d_in[7];
    const int*   nright   = (const int*)d_in[8];
    float*       out      = (float*)d_out;

    // Allow the large dynamic-LDS allocation (no-op if not required).
    (void)hipFuncSetAttribute((const void*)problog_scan_kernel,
                              hipFuncAttributeMaxDynamicSharedMemorySize,
                              SMEM_BYTES);

    const int grid = (B_ + C_ - 1) / C_;   // 342 workgroups, 12 columns each
    problog_scan_kernel<<<grid, THREADS_, SMEM_BYTES, stream>>>(
        imgs, ctl, ptl, leaf_var, leaf_cls, leaf_neg, gate, nleft, nright, out);
}